// GCN_78486232367723
// MI455X (gfx1250) — compile-verified
//
#include <hip/hip_runtime.h>
#include <hip/hip_bf16.h>

typedef __attribute__((ext_vector_type(2))) float v2f;
typedef __attribute__((ext_vector_type(8))) float v8f;

// ---------------------------------------------------------------------------
// WMMA fp32 GEMM:  H[M x N] = A[M x 128] * W[128 x N]   (K fixed at 128)
// One wave -> one 16x16 output tile. 8 waves/block share a 128x16 W panel in
// LDS. Uses V_WMMA_F32_16X16X4_F32 (exact fp32 path, per CDNA5 ISA 7.12).
// ---------------------------------------------------------------------------
__global__ __launch_bounds__(256) void gcn_gemm_wmma(
    const float* __restrict__ A, const float* __restrict__ W,
    float* __restrict__ H, int Mtiles, int N)
{
    __shared__ float Wlds[128 * 16];   // [k][n], 8 KB

    const int n0 = blockIdx.y * 16;
    // Cooperative load of 128x16 W panel (zero-pad cols >= N)
    for (int i = threadIdx.x; i < 128 * 16; i += 256) {
        const int k = i >> 4, nn = i & 15;
        float v = 0.0f;
        if (n0 + nn < N) v = W[k * N + n0 + nn];
        Wlds[i] = v;
    }
    __syncthreads();

    const int wave = threadIdx.x >> 5;
    const int lane = threadIdx.x & 31;
    const int mtile = blockIdx.x * 8 + wave;
    if (mtile >= Mtiles) return;                 // wave-uniform: EXEC stays full

    const int mlo   = lane & 15;
    const int khalf = (lane >> 4) << 1;          // 0 (lanes 0-15) or 2 (lanes 16-31)
    const float* __restrict__ arow = A + (long long)(mtile * 16 + mlo) * 128;

    v8f c = {};
    #pragma unroll 8
    for (int k = 0; k < 128; k += 4) {
        const int kk = k + khalf;
        // A frag (16x4): lane m = lane&15 holds K = kk, kk+1  (8B aligned)
        v2f a = *(const v2f*)(arow + kk);
        // B frag (4x16): row kk (lanes 0-15) / row kk+2 (lanes 16-31), col = lane&15
        v2f b;
        b.x = Wlds[kk * 16 + mlo];
        b.y = Wlds[(kk + 1) * 16 + mlo];
        c = __builtin_amdgcn_wmma_f32_16x16x4_f32(
                /*neg_a=*/false, a, /*neg_b=*/false, b,
                /*c_mod=*/(short)0, c, /*reuse_a=*/false, /*reuse_b=*/false);
    }

    // C/D layout: VGPR v -> row (v + 8*(lane>=16)), col = lane&15
    const int row0 = mtile * 16 + ((lane >> 4) << 3);
    const int col  = n0 + mlo;
    if (col < N) {
        #pragma unroll
        for (int v = 0; v < 8; ++v)
            H[(long long)(row0 + v) * N + col] = c[v];
    }
}

// ---------------------------------------------------------------------------
// Degree / normalization (deg = |dst==i| + 1 self loop; dinv = rsqrt(deg))
// ---------------------------------------------------------------------------
__global__ void gcn_deg_init(int* __restrict__ deg, int n) {
    int i = blockIdx.x * blockDim.x + threadIdx.x;
    if (i < n) deg[i] = 1;                       // self loop
}
__global__ void gcn_deg_count(const int* __restrict__ dst, int* __restrict__ deg, long long E) {
    long long e = (long long)blockIdx.x * blockDim.x + threadIdx.x;
    if (e < E) atomicAdd(&deg[dst[e]], 1);
}
__global__ void gcn_dinv(const int* __restrict__ deg, float* __restrict__ dinv, int n) {
    int i = blockIdx.x * blockDim.x + threadIdx.x;
    if (i < n) dinv[i] = rsqrtf((float)deg[i]);
}

// ---------------------------------------------------------------------------
// out[i,f] = bias[f] + dinv[i]^2 * h[i,f]     (self-loop term + bias)
// ---------------------------------------------------------------------------
__global__ void gcn_init_out(const float* __restrict__ h, const float* __restrict__ bias,
                             const float* __restrict__ dinv, float* __restrict__ out,
                             long long n, int F)
{
    long long idx = (long long)blockIdx.x * blockDim.x + threadIdx.x;
    if (idx >= n * F) return;
    const int i = (int)(idx / F);
    const int f = (int)(idx % F);
    const float di = dinv[i];
    out[idx] = bias[f] + di * di * h[idx];
}

// ---------------------------------------------------------------------------
// Edge scatter: out[dst] += h[src] * dinv[src]*dinv[dst]
// F=128 path: 32 lanes/edge, float4 per lane -> global_atomic_add_f32 at L2.
// ---------------------------------------------------------------------------
__global__ void gcn_scatter128(const float* __restrict__ h, const int* __restrict__ src,
                               const int* __restrict__ dst, const float* __restrict__ dinv,
                               float* __restrict__ out, long long E)
{
    long long t = (long long)blockIdx.x * blockDim.x + threadIdx.x;
    if (t >= E * 32) return;
    const long long e = t >> 5;
    const int q = (int)(t & 31);
    const int s = src[e], d = dst[e];
    const float norm = dinv[s] * dinv[d];
    const float4 hv = *(const float4*)(h + (long long)s * 128 + q * 4);
    float* op = out + (long long)d * 128 + q * 4;
    unsafeAtomicAdd(op + 0, hv.x * norm);
    unsafeAtomicAdd(op + 1, hv.y * norm);
    unsafeAtomicAdd(op + 2, hv.z * norm);
    unsafeAtomicAdd(op + 3, hv.w * norm);
}

// Generic scalar scatter (used for F=10 output layer)
__global__ void gcn_scatterF(const float* __restrict__ h, const int* __restrict__ src,
                             const int* __restrict__ dst, const float* __restrict__ dinv,
                             float* __restrict__ out, long long E, int F)
{
    long long t = (long long)blockIdx.x * blockDim.x + threadIdx.x;
    if (t >= E * F) return;
    const long long e = t / F;
    const int f = (int)(t % F);
    const int s = src[e], d = dst[e];
    const float norm = dinv[s] * dinv[d];
    unsafeAtomicAdd(out + (long long)d * F + f, h[(long long)s * F + f] * norm);
}

__global__ void gcn_relu(float* __restrict__ x, long long nelem) {
    long long i = (long long)blockIdx.x * blockDim.x + threadIdx.x;
    if (i < nelem) x[i] = fmaxf(x[i], 0.0f);
}

// ---------------------------------------------------------------------------
// Global mean pool + log_softmax
// ---------------------------------------------------------------------------
__global__ void pool_zero(float* __restrict__ sums, float* __restrict__ cnts, int ngraphs, int C) {
    int i = blockIdx.x * blockDim.x + threadIdx.x;
    if (i < ngraphs * C) sums[i] = 0.0f;
    if (i < ngraphs) cnts[i] = 0.0f;
}
__global__ void pool_accum(const float* __restrict__ h, const int* __restrict__ batch,
                           float* __restrict__ sums, float* __restrict__ cnts, int n, int C)
{
    int i = blockIdx.x * blockDim.x + threadIdx.x;
    if (i >= n) return;
    const int g = batch[i];
    for (int c = 0; c < 10; ++c)
        unsafeAtomicAdd(&sums[g * C + c], h[(long long)i * C + c]);
    unsafeAtomicAdd(&cnts[g], 1.0f);
}
__global__ void pool_final(const float* __restrict__ sums, const float* __restrict__ cnts,
                           float* __restrict__ out, int ngraphs, int C)
{
    int g = blockIdx.x * blockDim.x + threadIdx.x;
    if (g >= ngraphs) return;
    const float cnt = fmaxf(cnts[g], 1.0f);
    float v[10];
    float m = -3.402823466e+38f;
    for (int c = 0; c < 10; ++c) { v[c] = sums[g * C + c] / cnt; m = fmaxf(m, v[c]); }
    float lse = 0.0f;
    for (int c = 0; c < 10; ++c) lse += __expf(v[c] - m);
    lse = __logf(lse);
    for (int c = 0; c < 10; ++c) out[g * C + c] = v[c] - m - lse;
}

// ---------------------------------------------------------------------------
extern "C" void kernel_launch(void* const* d_in, const int* in_sizes, int n_in,
                              void* d_out, int out_size, void* d_ws, size_t ws_size,
                              hipStream_t stream)
{
    const float* x     = (const float*)d_in[0];
    const int*   edge  = (const int*)  d_in[1];
    const int*   batch = (const int*)  d_in[2];
    const float* Wmat[4] = { (const float*)d_in[3], (const float*)d_in[5],
                             (const float*)d_in[7], (const float*)d_in[9] };
    const float* bvec[4] = { (const float*)d_in[4], (const float*)d_in[6],
                             (const float*)d_in[8], (const float*)d_in[10] };

    const int       n = in_sizes[0] / 128;         // 100000 nodes
    const long long E = (long long)in_sizes[1] / 2; // 3,200,000 edges
    const int*  src = edge;
    const int*  dst = edge + E;
    const int   NG = 512, C = 10, H = 128;
    const int   Mtiles = (n + 15) / 16;            // 6250

    // ---- workspace carve-up ----
    float* bufA = (float*)d_ws;                        // n*128 activations
    float* bufH = bufA + (long long)n * H;             // n*128 gemm out
    float* dinv = bufH + (long long)n * H;             // n
    int*   deg  = (int*)(dinv + n);                    // n
    float* out3 = (float*)(deg + n);                   // n*10
    float* sums = out3 + (long long)n * C;             // 512*10
    float* cnts = sums + NG * C;                       // 512

    const int TB = 256;
    // ---- degree / dinv (identical for every layer) ----
    gcn_deg_init <<<(n + TB - 1) / TB, TB, 0, stream>>>(deg, n);
    gcn_deg_count<<<(unsigned)((E + TB - 1) / TB), TB, 0, stream>>>(dst, deg, E);
    gcn_dinv     <<<(n + TB - 1) / TB, TB, 0, stream>>>(deg, dinv, n);

    const dim3 gemmGrid((Mtiles + 7) / 8, H / 16);
    const long long nH = (long long)n * H;

    // ---- 3 hidden GCN layers (GEMM -> init(bias+self) -> scatter -> relu) ----
    for (int l = 0; l < 3; ++l) {
        const float* act = (l == 0) ? x : bufA;
        gcn_gemm_wmma<<<gemmGrid, TB, 0, stream>>>(act, Wmat[l], bufH, Mtiles, H);
        gcn_init_out <<<(unsigned)((nH + TB - 1) / TB), TB, 0, stream>>>(bufH, bvec[l], dinv, bufA, n, H);
        gcn_scatter128<<<(unsigned)((E * 32 + TB - 1) / TB), TB, 0, stream>>>(bufH, src, dst, dinv, bufA, E);
        gcn_relu     <<<(unsigned)((nH + TB - 1) / TB), TB, 0, stream>>>(bufA, nH);
    }

    // ---- output layer (N = 10, no relu) ----
    gcn_gemm_wmma<<<dim3((Mtiles + 7) / 8, 1), TB, 0, stream>>>(bufA, Wmat[3], bufH, Mtiles, C);
    gcn_init_out <<<(unsigned)(((long long)n * C + TB - 1) / TB), TB, 0, stream>>>(bufH, bvec[3], dinv, out3, n, C);
    gcn_scatterF <<<(unsigned)((E * C + TB - 1) / TB), TB, 0, stream>>>(bufH, src, dst, dinv, out3, E, C);

    // ---- mean pool + log_softmax ----
    pool_zero  <<<(NG * C + TB - 1) / TB, TB, 0, stream>>>(sums, cnts, NG, C);
    pool_accum <<<(n + TB - 1) / TB, TB, 0, stream>>>(out3, batch, sums, cnts, n, C);
    pool_final <<<(NG + TB - 1) / TB, TB, 0, stream>>>(sums, cnts, (float*)d_out, NG, C);
}